// TE_55044300865689
// MI455X (gfx1250) — compile-verified
//
#include <hip/hip_runtime.h>
#include <stdint.h>

typedef __attribute__((ext_vector_type(4))) unsigned int v4u;
typedef __attribute__((ext_vector_type(8))) int v8i;
typedef __attribute__((ext_vector_type(4))) int v4i;

#define RR 16
#define SPK 128
#define TAU_INV (1.0f / 50.0f)

// uniform-lane broadcast via v_readlane_b32 (no LDS round trip, unlike __shfl)
__device__ __forceinline__ int readlane_i(int v, int l) {
    return __builtin_amdgcn_readlane(v, l);
}
__device__ __forceinline__ float readlane_f(float v, int l) {
    return __builtin_bit_cast(float, __builtin_amdgcn_readlane(__builtin_bit_cast(int, v), l));
}

// ---------------------------------------------------------------------------
// Zero the dense output (37.7MB) at HBM write roofline with b128 stores.
// ---------------------------------------------------------------------------
__global__ void te_zero(float* __restrict__ out, int n) {
    int i = blockIdx.x * blockDim.x + threadIdx.x;
    int n4 = n >> 2;
    if (i < n4) {
        reinterpret_cast<float4*>(out)[i] = make_float4(0.f, 0.f, 0.f, 0.f);
    }
    if (i == 0) {
        for (int t = n & ~3; t < n; ++t) out[t] = 0.f;
    }
}

// ---------------------------------------------------------------------------
// One wave32 per sample. Since ttp_{n+1} == ttc_n in the reference scan, the
// whole 127-step chain reduces to ONE running grid, and that grid is nonzero
// only at the <=128 event pixels -> hold it as 4 floats/lane in VGPRs.
// Events are staged into LDS via the Tensor Data Mover (async, TENSORcnt).
// Only each slot's FINAL snapshot (found via LDS ds_max) is scattered.
// ---------------------------------------------------------------------------
__global__ __launch_bounds__(32) void te_chain(const float* __restrict__ event,
                                               const int* __restrict__ ttrace,
                                               const int* __restrict__ length,
                                               const int* __restrict__ testp,
                                               float* __restrict__ out) {
    const int s = blockIdx.x;
    const int lane = threadIdx.x;

    __shared__ float evt[SPK * 4];   // TDM destination: 128 events x (x,y,p,t)
    __shared__ int lastw[RR];        // last step that wrote each slot

    // ---- TDM: async-copy this sample's 2KB event tile into LDS ----
    unsigned long long ga = (unsigned long long)(uintptr_t)(event + (size_t)s * SPK * 4);
    unsigned lds_off = (unsigned)(uintptr_t)&evt[0]; // low 32 bits of flat addr = LDS offset
    v4u g0; v8i g1; v4i g2, g3;
    g0[0] = 1u;                                                // count=1, user mode
    g0[1] = lds_off;                                           // lds_addr
    g0[2] = (unsigned)(ga & 0xFFFFFFFFu);                      // global_addr[31:0]
    g0[3] = (unsigned)((ga >> 32) & 0x1FFFFFFu) | (2u << 30);  // global_addr[56:32], type=2
    g1[0] = (2 << 16);            // workgroup_mask=0 (not in cluster), data_size=4B
    g1[1] = (SPK * 4) << 16;      // tensor_dim0 = 512 elements (lo16)
    g1[2] = (1 << 16);            // tensor_dim1 = 1
    g1[3] = (SPK * 4) << 16;      // tile_dim0 = 512
    g1[4] = 1;                    // tile_dim1 = 1, tile_dim2 unused
    g1[5] = SPK * 4;              // tensor_dim0_stride = 512
    g1[6] = 0;
    g1[7] = 0;
    g2[0] = 1; g2[1] = 1; g2[2] = SPK * 4; g2[3] = 0;  // dims 2/3 degenerate
    g3[0] = 0; g3[1] = 0; g3[2] = 0; g3[3] = 0;
#if defined(__clang_major__) && __clang_major__ >= 23
    v8i g4 = {0, 0, 0, 0, 0, 0, 0, 0};
    __builtin_amdgcn_tensor_load_to_lds(g0, g1, g2, g3, g4, 0);
#else
    __builtin_amdgcn_tensor_load_to_lds(g0, g1, g2, g3, 0);
#endif

    // ---- work independent of the event tile (overlaps TDM latency) ----
    if (lane < RR) lastw[lane] = -1;
    int ttv[4];
#pragma unroll
    for (int c = 0; c < 4; ++c)
        ttv[c] = ttrace[(size_t)s * SPK + c * 32 + lane];
    const int len = length[s];
    const int test = testp[0];
    const int side = (test == 1) ? 76 : 96;
    const int off = (128 - side) / 2;

    // last writer per slot over steps n=1..127 (same-wave DS ops are in-order)
#pragma unroll
    for (int c = 0; c < 4; ++c) {
        int n = c * 32 + lane;
        if (n > 0) atomicMax(&lastw[ttv[c] & (RR - 1)], n);
    }

    __builtin_amdgcn_s_wait_tensorcnt(0);  // event tile now resident in LDS

    // per-lane step parameters for n = c*32 + lane
    float mmv[4];
    int keyv[4], ctrlv[4];
#pragma unroll
    for (int c = 0; c < 4; ++c) {
        int n = c * 32 + lane;
        float fx = evt[n * 4 + 0], fy = evt[n * 4 + 1];
        float fp = evt[n * 4 + 2], ft = evt[n * 4 + 3];
        keyv[c] = (((int)fp) << 14) | (((int)fx) << 7) | ((int)fy);
        float tp = (n > 0) ? evt[(n - 1) * 4 + 3] : ft;
        mmv[c] = __expf((tp - ft) * TAU_INV);  // n==0 -> exp(0)=1 (no-op step)
        bool alive = (n > 0) && (len >= n);
        bool islast = (n > 0) && (lastw[ttv[c] & (RR - 1)] == n);
        ctrlv[c] = (ttv[c] & (RR - 1)) | (alive ? 16 : 0) | (islast ? 32 : 0);
    }

    // initial running grid = slot tt[0]'s content (impulse only if tt[0]==0)
    int tt0 = readlane_i(ttv[0], 0);
    int key0 = readlane_i(keyv[0], 0);
    float row[4];
#pragma unroll
    for (int j = 0; j < 4; ++j)
        row[j] = (tt0 == 0 && keyv[j] == key0) ? 0.5f : 0.0f;

    const size_t outbase_s = (size_t)s * RR * 2 * side * side;

    // ---- the 127-step chain (v_readlane broadcasts on the critical path) ----
    for (int c = 0; c < 4; ++c) {
        for (int k = 0; k < 32; ++k) {
            int ctrl   = readlane_i(ctrlv[c], k);
            float MM   = readlane_f(mmv[c], k);
            int keyn   = readlane_i(keyv[c], k);
            float pre  = readlane_f(row[c], k);       // grid value at key_n (pre-decay)
            float add  = (ctrl & 16) ? 0.5f * (1.0f - MM * pre) : 0.0f;
#pragma unroll
            for (int j = 0; j < 4; ++j)
                row[j] = MM * row[j] + ((keyv[j] == keyn) ? add : 0.0f);
            if (ctrl & 32) {  // final snapshot for slot r -> scatter (uniform branch)
                int r = ctrl & (RR - 1);
                float* dst = out + outbase_s + (size_t)r * 2 * side * side;
#pragma unroll
                for (int j = 0; j < 4; ++j) {
                    int kj = keyv[j];
                    int yy = kj & 127, xx = (kj >> 7) & 127, pp = (kj >> 14) & 1;
                    unsigned ux = (unsigned)(xx - off), uy = (unsigned)(yy - off);
                    if (ux < (unsigned)side && uy < (unsigned)side)
                        dst[((size_t)pp * side + ux) * side + uy] = row[j];
                }
            }
        }
    }

    // slot 0 never overwritten -> the t=0 impulse survives
    if (lane == 0 && lastw[0] < 0) {
        int yy = key0 & 127, xx = (key0 >> 7) & 127, pp = (key0 >> 14) & 1;
        unsigned ux = (unsigned)(xx - off), uy = (unsigned)(yy - off);
        if (ux < (unsigned)side && uy < (unsigned)side)
            out[outbase_s + ((size_t)pp * side + ux) * side + uy] = 0.5f;
    }
}

extern "C" void kernel_launch(void* const* d_in, const int* in_sizes, int n_in,
                              void* d_out, int out_size, void* d_ws, size_t ws_size,
                              hipStream_t stream) {
    const float* event = (const float*)d_in[0];
    const int* ttrace  = (const int*)d_in[1];
    const int* length  = (const int*)d_in[2];
    const int* testp   = (const int*)d_in[3];
    float* out = (float*)d_out;

    const int SN = in_sizes[2];  // number of samples (length has SN entries)

    int n4 = out_size >> 2;
    int zblocks = (n4 + 255) / 256;
    if (zblocks < 1) zblocks = 1;
    te_zero<<<zblocks, 256, 0, stream>>>(out, out_size);
    te_chain<<<SN, 32, 0, stream>>>(event, ttrace, length, testp, out);
}